// CrossAttention_34565896798833
// MI455X (gfx1250) — compile-verified
//
#include <hip/hip_runtime.h>
#include <hip/hip_bf16.h>
#include <math.h>

// ---------------------------------------------------------------------------
// Criss-cross attention forward for MI455X (gfx1250, wave32, WMMA).
// All matmul stages use V_WMMA_F32_16X16X4_F32 (full fp32 precision).
//
// Problem constants (from reference):
//   B=16, C=256, N=128 (H=W), O=2*CQK+C=512, CQK=128, P=N*N=16384 pixels.
//
// Workspace layout (floats), total ~514 MB (assumed available in d_ws):
//   sA  [B*C]          per-(b,c) scale      (rsqrt(var+eps)*w)
//   sB  [B*C]          per-(b,c) shift      (bias - mu*scale)
//   Wt  [B*O]          qkv_w @ sB           (bias folded through GEMM)
//   nrm [2*B*N*N]      softmax (rowmax, 1/rowsum)
//   qkv [B*O*P]        q (o<128), k (128..255), v (256..511)
// ---------------------------------------------------------------------------

#define B_    16
#define C_    256
#define N_    128
#define O_    512
#define CQK_  128
#define P_    (N_ * N_)
#define EPS_  1e-5f
#define NEG_FILL_ (-1e20f)

typedef __attribute__((ext_vector_type(2))) float v2f;
typedef __attribute__((ext_vector_type(8))) float v8f;

// D(16x16,f32) = A(16x4,f32) * B(4x16,f32) + C  — wave32 matrix op.
__device__ __forceinline__ v8f wmma4(v2f a, v2f b, v8f c) {
  return __builtin_amdgcn_wmma_f32_16x16x4_f32(
      /*neg_a=*/false, a, /*neg_b=*/false, b,
      /*c_mod=*/(short)0, c, /*reuse_a=*/false, /*reuse_b=*/false);
}

// ---------------------------------------------------------------------------
// K1: per-(b,c) mean/var over 16384 elems -> affine (sA, sB).
// grid = B*C blocks x 256 threads
// ---------------------------------------------------------------------------
__global__ void stats_kernel(const float* __restrict__ x,
                             const float* __restrict__ in_w,
                             const float* __restrict__ in_b,
                             float* __restrict__ sA,
                             float* __restrict__ sB) {
  const int bc = blockIdx.x;           // b*C + c
  const int c  = bc % C_;
  const float* xp = x + (size_t)bc * P_;
  float s = 0.f, s2 = 0.f;
  for (int i = threadIdx.x; i < P_; i += blockDim.x) {
    float v = xp[i];
    s += v;
    s2 += v * v;
  }
  __shared__ float r0[256];
  __shared__ float r1[256];
  r0[threadIdx.x] = s;
  r1[threadIdx.x] = s2;
  __syncthreads();
  for (int off = 128; off > 0; off >>= 1) {
    if ((int)threadIdx.x < off) {
      r0[threadIdx.x] += r0[threadIdx.x + off];
      r1[threadIdx.x] += r1[threadIdx.x + off];
    }
    __syncthreads();
  }
  if (threadIdx.x == 0) {
    const float mu  = r0[0] * (1.0f / P_);
    const float var = r1[0] * (1.0f / P_) - mu * mu;
    const float rs  = rsqrtf(var + EPS_);
    const float w   = in_w[c];
    sA[bc] = rs * w;
    sB[bc] = in_b[c] - mu * rs * w;
  }
}

// ---------------------------------------------------------------------------
// K1b: Wt[b,o] = sum_c qkv_w[o,c] * sB[b,c].  grid = B*O/256 x 256
// ---------------------------------------------------------------------------
__global__ void wt_kernel(const float* __restrict__ qkv_w,
                          const float* __restrict__ sB,
                          float* __restrict__ Wt) {
  const int idx = blockIdx.x * blockDim.x + threadIdx.x;  // b*O + o
  const int b = idx / O_;
  const int o = idx % O_;
  const float* wr = qkv_w + (size_t)o * C_;
  const float* t  = sB + b * C_;
  float acc = 0.f;
  for (int c = 0; c < C_; ++c) acc += wr[c] * t[c];
  Wt[idx] = acc;
}

// ---------------------------------------------------------------------------
// K2: qkv[b,o,p] = sum_c (qkv_w[o,c]*sA[b,c]) * x[b,c,p] + Wt[b,o]
// GEMM M=512, N=16384, K=256 per batch via f32 WMMA.
// block = 128 thr (4 waves); wave tile = 16 o x 64 px (A reused over 4 B).
// grid = (P/64, O/64, B)
// ---------------------------------------------------------------------------
__global__ void qkv_gemm_kernel(const float* __restrict__ x,
                                const float* __restrict__ qkv_w,
                                const float* __restrict__ sA,
                                const float* __restrict__ Wt,
                                float* __restrict__ qkv) {
  const int lane = threadIdx.x & 31;
  const int wave = threadIdx.x >> 5;
  const int lm   = lane & 15;      // A row / B col / D col
  const int half = lane >> 4;      // k sub-pair selector
  const int b     = blockIdx.z;
  const int oBase = blockIdx.y * 64 + wave * 16;
  const int pBase = blockIdx.x * 64;
  const float* xb  = x + (size_t)b * C_ * P_;
  const float* sAb = sA + b * C_;

  v8f acc[4] = {};
  for (int c = 0; c < C_; c += 4) {
    const int ca = c + 2 * half;
    const float* wr = qkv_w + (size_t)(oBase + lm) * C_ + ca;
    v2f a;
    a.x = wr[0];
    a.y = wr[1];
    const float s0 = sAb[ca];
    const float s1 = sAb[ca + 1];
    const float* x0 = xb + (size_t)ca * P_;
    const float* x1 = x0 + P_;
#pragma unroll
    for (int t = 0; t < 4; ++t) {
      const int p = pBase + t * 16 + lm;
      v2f bb;
      bb.x = x0[p] * s0;
      bb.y = x1[p] * s1;
      acc[t] = wmma4(a, bb, acc[t]);
    }
  }

  float* ob = qkv + (size_t)b * O_ * P_;
#pragma unroll
  for (int r = 0; r < 8; ++r) {
    const int o = oBase + r + 8 * half;
    const float wt = Wt[b * O_ + o];
    float* orow = ob + (size_t)o * P_ + pBase + lm;
#pragma unroll
    for (int t = 0; t < 4; ++t) orow[t * 16] = acc[t][r] + wt;
  }
}

// ---------------------------------------------------------------------------
// K3: energy. dir=0 (H): fixed (b,w), E[h,g]=sum_c q[c,h,w]*k[c,g,w], diag=-inf
//     dir=1 (W): fixed (b,h), E[w,v]=sum_c q[c,h,w]*k[c,h,v]
// mask==0 -> NEG_FILL (applied last, like the reference).
// block = 128 thr (4 waves); wave = 16m x 64n; grid = (4, N, 2*B)
// ---------------------------------------------------------------------------
__global__ void energy_kernel(const float* __restrict__ qkv,
                              const int* __restrict__ mask,
                              float* __restrict__ energy) {
  const int lane = threadIdx.x & 31;
  const int wave = threadIdx.x >> 5;
  const int lm   = lane & 15;
  const int half = lane >> 4;
  const int dir = blockIdx.z & 1;
  const int b   = blockIdx.z >> 1;
  const int fix = blockIdx.y;                 // w if dir==0, h if dir==1
  const int tid = blockIdx.x * 4 + wave;      // 0..15
  const int mBase = (tid & 7) * 16;
  const int nBase = (tid >> 3) * 64;
  const int sI = dir ? 1 : N_;                // pixel stride along varying dim
  const int bF = dir ? fix * N_ : fix;        // fixed-dim pixel base
  const float* qb = qkv + (size_t)b * O_ * P_;
  const float* kb = qb + (size_t)CQK_ * P_;

  v8f acc[4] = {};
  const int pixA = bF + (mBase + lm) * sI;
  for (int c = 0; c < CQK_; c += 4) {
    const int ca = c + 2 * half;
    const float* q0 = qb + (size_t)ca * P_;
    const float* k0 = kb + (size_t)ca * P_;
    v2f a;
    a.x = q0[pixA];
    a.y = q0[P_ + pixA];
#pragma unroll
    for (int t = 0; t < 4; ++t) {
      const int pixB = bF + (nBase + t * 16 + lm) * sI;
      v2f bb;
      bb.x = k0[pixB];
      bb.y = k0[P_ + pixB];
      acc[t] = wmma4(a, bb, acc[t]);
    }
  }

#pragma unroll
  for (int r = 0; r < 8; ++r) {
    const int mrow = mBase + r + 8 * half;    // h (dir 0) or w (dir 1)
    const int h = dir ? fix : mrow;
    const int w = dir ? mrow : fix;
    const int mk = mask[(b * N_ + h) * N_ + w];
    float* erow = energy + ((size_t)(b * N_ + h) * N_ + w) * (2 * N_)
                  + (dir ? N_ : 0);
#pragma unroll
    for (int t = 0; t < 4; ++t) {
      const int g = nBase + t * 16 + lm;
      float e = acc[t][r];
      if (dir == 0 && g == mrow) e = -__builtin_inff();  // diag mask (H only)
      if (mk == 0) e = NEG_FILL_;                        // spatial mask last
      erow[g] = e;
    }
  }
}

// ---------------------------------------------------------------------------
// K4: per-row softmax normalizers over 256 logits: nrm = {rowmax, 1/rowsum}.
// one wave per row; grid = (B*N*N)/4 blocks x 128 threads.
// ---------------------------------------------------------------------------
__global__ void softmax_norm_kernel(const float* __restrict__ energy,
                                    float* __restrict__ nrm) {
  const int row  = blockIdx.x * (blockDim.x >> 5) + (threadIdx.x >> 5);
  const int lane = threadIdx.x & 31;
  const float* er = energy + (size_t)row * (2 * N_);
  float v[8];
  float mx = -__builtin_inff();
#pragma unroll
  for (int i = 0; i < 8; ++i) {
    v[i] = er[lane + 32 * i];
    mx = fmaxf(mx, v[i]);
  }
#pragma unroll
  for (int off = 16; off > 0; off >>= 1) mx = fmaxf(mx, __shfl_xor(mx, off, 32));
  float s = 0.f;
#pragma unroll
  for (int i = 0; i < 8; ++i) s += expf(v[i] - mx);
#pragma unroll
  for (int off = 16; off > 0; off >>= 1) s += __shfl_xor(s, off, 32);
  if (lane == 0) {
    nrm[2 * row]     = mx;
    nrm[2 * row + 1] = 1.0f / s;
  }
}

// ---------------------------------------------------------------------------
// K5: out contribution.  dir=0: out = x + V * attH   (per (b,w) column GEMM)
//                        dir=1: out += V * attW      (per (b,h) row GEMM)
// block = 512 thr (16 waves = all 256 channels); att tile (128g x 16n) is
// exponentiated ONCE into LDS, then consumed as WMMA B fragments.
// grid = (N/16, N, B); launched twice (dir=0 then dir=1).
// ---------------------------------------------------------------------------
__global__ void out_kernel(const float* __restrict__ qkv,
                           const float* __restrict__ energy,
                           const float* __restrict__ nrm,
                           const float* __restrict__ x,
                           float* __restrict__ out,
                           int dir) {
  __shared__ float attS[N_ * 16];   // attS[g*16 + n]
  const int b   = blockIdx.z;
  const int fix = blockIdx.y;       // w (dir 0) or h (dir 1)
  const int nb  = blockIdx.x * 16;  // tile base along varying dim

  {  // stage softmaxed attention tile into LDS (32 thr per row, 4 g each)
    const int n  = threadIdx.x >> 5;
    const int gl = threadIdx.x & 31;
    const int rowIdx = dir ? ((b * N_ + fix) * N_ + nb + n)
                           : ((b * N_ + nb + n) * N_ + fix);
    const float mx  = nrm[2 * rowIdx];
    const float inv = nrm[2 * rowIdx + 1];
    const float* er = energy + (size_t)rowIdx * (2 * N_) + (dir ? N_ : 0);
#pragma unroll
    for (int i = 0; i < 4; ++i) {
      const int g = gl + 32 * i;
      attS[g * 16 + n] = expf(er[g] - mx) * inv;
    }
  }
  __syncthreads();

  const int lane = threadIdx.x & 31;
  const int wave = threadIdx.x >> 5;
  const int lm   = lane & 15;
  const int half = lane >> 4;
  const int cB   = wave * 16;               // 16 waves cover c=0..255
  const int sI = dir ? 1 : N_;
  const int bF = dir ? fix * N_ : fix;
  const float* vrow = qkv + ((size_t)b * O_ + 2 * CQK_ + cB + lm) * P_ + bF;

  v8f acc = {};
  for (int g = 0; g < N_; g += 4) {
    const int ga = g + 2 * half;
    v2f a, bb;
    a.x  = vrow[(size_t)ga * sI];
    a.y  = vrow[(size_t)(ga + 1) * sI];
    bb.x = attS[ga * 16 + lm];
    bb.y = attS[(ga + 1) * 16 + lm];
    acc = wmma4(a, bb, acc);
  }

#pragma unroll
  for (int r = 0; r < 8; ++r) {
    const int c = cB + r + 8 * half;
    const size_t oIdx =
        ((size_t)b * C_ + c) * P_ + bF + (size_t)(nb + lm) * sI;
    const float base = dir ? out[oIdx] : x[oIdx];
    out[oIdx] = base + acc[r];
  }
}

// ---------------------------------------------------------------------------
extern "C" void kernel_launch(void* const* d_in, const int* in_sizes, int n_in,
                              void* d_out, int out_size, void* d_ws,
                              size_t ws_size, hipStream_t stream) {
  (void)in_sizes; (void)n_in; (void)out_size; (void)ws_size;
  const float* x      = (const float*)d_in[0];
  const int*   mask   = (const int*)d_in[1];
  const float* in_w   = (const float*)d_in[2];
  const float* in_b   = (const float*)d_in[3];
  const float* qkv_w  = (const float*)d_in[4];

  float* out    = (float*)d_out;                       // (B,C,N,N)
  float* energy = out + (size_t)B_ * C_ * P_;          // (B,N,N,2N)

  float* ws  = (float*)d_ws;
  float* sA  = ws;                                     // B*C
  float* sB  = sA + B_ * C_;                           // B*C
  float* Wt  = sB + B_ * C_;                           // B*O
  float* nrm = Wt + B_ * O_;                           // 2*B*N*N
  float* qkv = nrm + 2 * (size_t)B_ * N_ * N_;         // B*O*P

  // 1) instance-norm affine coefficients
  stats_kernel<<<B_ * C_, 256, 0, stream>>>(x, in_w, in_b, sA, sB);
  // 1b) bias folded through the projection
  wt_kernel<<<(B_ * O_) / 256, 256, 0, stream>>>(qkv_w, sB, Wt);
  // 2) QKV projection GEMM (WMMA f32)
  qkv_gemm_kernel<<<dim3(P_ / 64, O_ / 64, B_), 128, 0, stream>>>(
      x, qkv_w, sA, Wt, qkv);
  // 3) energy (H and W branches) + diag/mask, written to output buffer
  energy_kernel<<<dim3(4, N_, 2 * B_), 128, 0, stream>>>(qkv, mask, energy);
  // 4) softmax row normalizers
  softmax_norm_kernel<<<(B_ * N_ * N_) / 4, 128, 0, stream>>>(energy, nrm);
  // 5) out = x + V*attH, then out += V*attW
  out_kernel<<<dim3(N_ / 16, N_, B_), 512, 0, stream>>>(qkv, energy, nrm, x,
                                                        out, 0);
  out_kernel<<<dim3(N_ / 16, N_, B_), 512, 0, stream>>>(qkv, energy, nrm, x,
                                                        out, 1);
}